// LCNSpiking2_28733331210638
// MI455X (gfx1250) — compile-verified
//
#include <hip/hip_runtime.h>

#define K_NN   25
#define BETA_C 0.85f

typedef __attribute__((ext_vector_type(2))) float v2f;
typedef __attribute__((ext_vector_type(8))) float v8f;

// One thread per (batch b, neuron d). Lanes sweep d -> coalesced record stores.
// h gathers are random within one batch row (L2-resident on MI455X's 192MB L2).
__global__ __launch_bounds__(256) void lcn_layer(
    const float* __restrict__ h, long hstride,            // input activations, row stride in elems
    const float* __restrict__ W, const float* __restrict__ bias,
    const int*   __restrict__ knn, const float* __restrict__ thr,
    float* __restrict__ mem_state,                        // (B,D) persistent, nullptr for last layer
    float* __restrict__ mem_out, float* __restrict__ spk_out,  // (B,D) record slices
    int D)
{
    const int d = blockIdx.x * 256 + threadIdx.x;
    const int b = blockIdx.y;
    const float* __restrict__ hrow = h + (long)b * hstride;
    const int base = d * K_NN;

    float acc = 0.0f;
#pragma unroll
    for (int k = 0; k < K_NN; ++k) {
        acc = __builtin_fmaf(hrow[knn[base + k]], W[base + k], acc);
    }

    const float cur = acc + bias[d];
    const float th  = thr[d];
    const long  od  = (long)b * D + d;

    float mem;
    if (mem_state != nullptr) {
        const float mp    = mem_state[od];
        const float reset = (mp > th) ? th : 0.0f;        // (mem_prev>thr)*thr folded
        mem = __builtin_fmaf(BETA_C, mp, cur) - reset;
        mem_state[od] = mem;
    } else {
        // Last layer: reference carries mem_prev (stuck at 0) => mem == cur, reset == 0.
        mem = cur;
    }

    mem_out[od] = mem;
    spk_out[od] = ((mem - th) > 0.0f) ? 1.0f : 0.0f;
}

// angles = h2(64x2048) @ Wfc.T(2048x2) + bfc, via V_WMMA_F32_16X16X4_F32.
// One wave per 16-row M tile; N padded 2->16 with zeroed B columns.
__global__ __launch_bounds__(32) void fc_wmma(
    const float* __restrict__ h2,   // (64, 2048) = mem_rec2 at t = T-1
    const float* __restrict__ Wfc,  // (2, 2048)
    const float* __restrict__ bfc,  // (2)
    float* __restrict__ ang)        // (64, 2)
{
    const int C     = 2048;
    const int lane  = threadIdx.x;
    const int half  = lane >> 4;            // A/B K-pair select (K = 2*half + {0,1})
    const int l     = lane & 15;            // A: row M; B: col N
    const int tileM = blockIdx.x * 16;

    const int   n    = (l < 2) ? l : 0;     // clamp so B addresses stay in-bounds
    const float keep = (l < 2) ? 1.0f : 0.0f;

    const float* __restrict__ arow = h2  + (long)(tileM + l) * C + 2 * half;
    const float* __restrict__ brow = Wfc + (long)n * C + 2 * half;

    v8f acc = {0.f, 0.f, 0.f, 0.f, 0.f, 0.f, 0.f, 0.f};
    for (int c = 0; c < C; c += 4) {
        v2f A, Bv;
        A.x  = arow[c];
        A.y  = arow[c + 1];
        Bv.x = brow[c]     * keep;          // zero padded columns (no OOB, no divergence)
        Bv.y = brow[c + 1] * keep;
        acc = __builtin_amdgcn_wmma_f32_16x16x4_f32(
            /*neg_a=*/false, A, /*neg_b=*/false, Bv,
            /*c_mod=*/(short)0, acc, /*reuse_a=*/false, /*reuse_b=*/false);
    }

    if (l < 2) {                            // divergence only after the last WMMA
        const float bv = bfc[l];
#pragma unroll
        for (int j = 0; j < 8; ++j) {
            const int row = tileM + half * 8 + j;   // C/D layout: VGPR j -> M = j (+8 for hi half)
            ang[row * 2 + l] = acc[j] + bv;
        }
    }
}

extern "C" void kernel_launch(void* const* d_in, const int* in_sizes, int n_in,
                              void* d_out, int out_size, void* d_ws, size_t ws_size,
                              hipStream_t stream) {
    (void)in_sizes; (void)n_in; (void)out_size; (void)ws_size;

    // setup_inputs() dict order: x, (W,b,knn,thr) x3, Wfc, bfc
    const float* x    = (const float*)d_in[0];
    const float* W0   = (const float*)d_in[1];
    const float* b0   = (const float*)d_in[2];
    const int*   knn0 = (const int*)  d_in[3];
    const float* thr0 = (const float*)d_in[4];
    const float* W1   = (const float*)d_in[5];
    const float* b1   = (const float*)d_in[6];
    const int*   knn1 = (const int*)  d_in[7];
    const float* thr1 = (const float*)d_in[8];
    const float* W2   = (const float*)d_in[9];
    const float* b2   = (const float*)d_in[10];
    const int*   knn2 = (const int*)  d_in[11];
    const float* thr2 = (const float*)d_in[12];
    const float* Wfc  = (const float*)d_in[13];
    const float* bfc  = (const float*)d_in[14];

    float* out = (float*)d_out;
    const int  B = 64, T = 20, Din = 16384;
    const int  D0 = 8192, D1 = 4096, D2 = 2048;

    // Output layout (flat, reference return order):
    const long m0 = 0;
    const long m1 = m0 + (long)T * B * D0;
    const long m2 = m1 + (long)T * B * D1;
    const long s0 = m2 + (long)T * B * D2;
    const long s1 = s0 + (long)T * B * D0;
    const long s2 = s1 + (long)T * B * D1;
    const long ao = s2 + (long)T * B * D2;

    // Persistent membrane state for layers 0 and 1 (layer 2's is identically zero).
    float* mem0 = (float*)d_ws;
    float* mem1 = mem0 + (long)B * D0;
    hipMemsetAsync(d_ws, 0, (size_t)(B * (D0 + D1)) * sizeof(float), stream);

    for (int t = 0; t < T; ++t) {
        // Layer 0: input is x[:, t, :] (row stride T*Din)
        lcn_layer<<<dim3(D0 / 256, B), 256, 0, stream>>>(
            x + (long)t * Din, (long)T * Din, W0, b0, knn0, thr0, mem0,
            out + m0 + (long)t * B * D0, out + s0 + (long)t * B * D0, D0);
        // Layer 1: input is this step's spk0 (already in d_out)
        lcn_layer<<<dim3(D1 / 256, B), 256, 0, stream>>>(
            out + s0 + (long)t * B * D0, (long)D0, W1, b1, knn1, thr1, mem1,
            out + m1 + (long)t * B * D1, out + s1 + (long)t * B * D1, D1);
        // Layer 2: input is this step's spk1; membrane state frozen at zero
        lcn_layer<<<dim3(D2 / 256, B), 256, 0, stream>>>(
            out + s1 + (long)t * B * D1, (long)D1, W2, b2, knn2, thr2, nullptr,
            out + m2 + (long)t * B * D2, out + s2 + (long)t * B * D2, D2);
    }

    // FC head on last step's layer-2 membrane (== cur2).
    fc_wmma<<<dim3(64 / 16), 32, 0, stream>>>(
        out + m2 + (long)(T - 1) * B * D2, Wfc, bfc, out + ao);
}